// Dynamics_20134806684008
// MI455X (gfx1250) — compile-verified
//
#include <hip/hip_runtime.h>
#include <hip/hip_bf16.h>
#include <math.h>

typedef __attribute__((ext_vector_type(2))) float v2f;
typedef __attribute__((ext_vector_type(8))) float v8f;

__device__ __forceinline__ float fast_tanh(float x) {
#if __has_builtin(__builtin_amdgcn_tanhf)
    return __builtin_amdgcn_tanhf(x);        // gfx1250 v_tanh_f32
#elif __has_builtin(__builtin_amdgcn_tanh_f32)
    return __builtin_amdgcn_tanh_f32(x);
#else
    return tanhf(x);
#endif
}

// One wave32 owns 16 columns (b values). T is processed in chunks of 16 steps.
// Within a chunk the inclusive prefix sum of DT*cos / DT*sin is computed as
//   D = C(carry) + sum_{g=0..3} A_g(16x4, lower-triangular slice) x B_g(4x16, values)
// via 4 chained v_wmma_f32_16x16x4_f32 per component (exact f32 accumulate).
//
// f32 16x4 A layout (ISA 7.12.2): lanes 0-15 & 16-31 hold rows M=0..15;
//   VGPR0 = K=0 (lanes 0-15) | K=2 (lanes 16-31), VGPR1 = K=1 | K=3.
// f32 4x16 B layout (row striped across lanes): VGPR0 = k=0 | k=2, VGPR1 = k=1 | k=3,
//   column n = lane & 15.
// f32 16x16 C/D layout: VGPR r: lanes 0-15 -> M=r, lanes 16-31 -> M=r+8, N = lane&15.
__global__ __launch_bounds__(256) void rollout_wmma_scan(
    const float* __restrict__ x0, const float* __restrict__ u,
    float* __restrict__ out, int Tsteps, int Bcols)
{
    const int lane = threadIdx.x & 31;
    const int wave = (int)(blockIdx.x * (blockDim.x >> 5)) + (threadIdx.x >> 5);
    const int half = lane >> 4;        // 0: lanes 0-15, 1: lanes 16-31
    const int n    = lane & 15;        // column within tile / row M of A

    const long B   = (long)Bcols;
    const long col = (long)wave * 16 + n;

    const float ANG = 0.52359877559829887308f;  // pi/6
    const float DT  = 0.1f;

    // Triangular A operand slices: A[m,k] = (k <= m) ? 1 : 0, constant over chunks.
    v2f amat[4];
#pragma unroll
    for (int g = 0; g < 4; ++g) {
        const int k0 = 4 * g + 2 * half;
        amat[g].x = (k0     <= n) ? 1.0f : 0.0f;
        amat[g].y = (k0 + 1 <= n) ? 1.0f : 0.0f;
    }

    // Carry = current position of this lane's column (both halves hold a copy).
    float cx = x0[col];
    float cy = x0[B + col];

    // Row t=0 of the output is x0 itself (one lane-half writes it).
    if (half == 0) {
        __builtin_nontemporal_store(cx, out + col);
        __builtin_nontemporal_store(cy, out + B + col);
    }

    const float* up = u + col;
    float*       op = out + col;

    for (int t0 = 0; t0 < Tsteps; t0 += 16) {
        v8f accx, accy;
#pragma unroll
        for (int r = 0; r < 8; ++r) { accx[r] = cx; accy[r] = cy; }

#pragma unroll
        for (int g = 0; g < 4; ++g) {
            const long k0 = (long)t0 + 4 * g + 2 * half;
            const float u0 = __builtin_nontemporal_load(up + k0 * B);
            const float u1 = __builtin_nontemporal_load(up + (k0 + 1) * B);
            const float a0 = fast_tanh(u0) * ANG;
            const float a1 = fast_tanh(u1) * ANG;
            v2f bc, bs;
            bc.x = __cosf(a0) * DT;  bc.y = __cosf(a1) * DT;
            bs.x = __sinf(a0) * DT;  bs.y = __sinf(a1) * DT;

            accx = __builtin_amdgcn_wmma_f32_16x16x4_f32(
                false, amat[g], false, bc, (short)0, accx, false, false);
            accy = __builtin_amdgcn_wmma_f32_16x16x4_f32(
                false, amat[g], false, bs, (short)0, accy, false, false);
        }

        // Prefetch next chunk of u into cache hierarchy (global_prefetch_b8).
        if (t0 + 16 < Tsteps) {
            __builtin_prefetch(up + ((long)t0 + 16 + 2 * half) * B, 0, 0);
        }

        // Store 16 output timesteps: lane-half 0 -> rows t0+1..t0+8, half 1 -> t0+9..t0+16.
#pragma unroll
        for (int r = 0; r < 8; ++r) {
            const long trow = (long)t0 + r + 8 * half + 1;
            __builtin_nontemporal_store(accx[r], op + trow * 2 * B);
            __builtin_nontemporal_store(accy[r], op + trow * 2 * B + B);
        }

        // New carry = D[M=15] = element 7 of upper lane half, same column.
        cx = __shfl(accx[7], 16 + n, 32);
        cy = __shfl(accy[7], 16 + n, 32);
    }
}

extern "C" void kernel_launch(void* const* d_in, const int* in_sizes, int n_in,
                              void* d_out, int out_size, void* d_ws, size_t ws_size,
                              hipStream_t stream) {
    const float* x0 = (const float*)d_in[0];     // [2, B]
    const float* u  = (const float*)d_in[1];     // [T, B]
    float* out = (float*)d_out;                  // [T+1, 2, B]

    const int B = in_sizes[0] / 2;
    const int T = in_sizes[1] / B;

    const int wavesPerBlock = 8;                 // 256 threads
    const int totalWaves = B / 16;               // 16 columns per wave
    const int blocks = (totalWaves + wavesPerBlock - 1) / wavesPerBlock;

    rollout_wmma_scan<<<blocks, wavesPerBlock * 32, 0, stream>>>(x0, u, out, T, B);
}